// NMS_Export_15728170238048
// MI455X (gfx1250) — compile-verified
//
#include <hip/hip_runtime.h>
#include <stdint.h>
#include <float.h>

// ---------------- problem constants (from reference) ----------------
#define NBATCH     8
#define NPRED      25200
#define NPAD       32768        // next pow2 >= NPRED, for bitonic sort
#define NCH        85
#define NCLS       80
#define TOPK       4096
#define MAX_DET    300
#define CONF_THR   0.001f
#define IOU_THR    0.45f
#define CLS_OFF    4096.0f

// ---------------- workspace layout (bytes) ----------------
#define WS_KEYS    0                               // u64 [8][32768]      = 2,097,152
#define WS_CLS     2097152                         // u8  [8][25200]      =   201,600
#define WS_OBOX    2298752                         // f4  [8][4096]       =   524,288 (16B aligned)
#define WS_DET     2823040                         // f32 [8][4096][6]    =   786,432
#define WS_VAL     3609472                         // u8  [8][4096]       =    32,768
// total ~3.64 MB

__device__ __forceinline__ unsigned orderable(float f) {
    unsigned u = __float_as_uint(f);
    return (u & 0x80000000u) ? ~u : (u | 0x80000000u);
}
__device__ __forceinline__ float inv_orderable(unsigned o) {
    unsigned ub = (o & 0x80000000u) ? (o & 0x7FFFFFFFu) : ~o;
    return __uint_as_float(ub);
}

// ---------------------------------------------------------------
// K1: per-row obj/conf/argmax -> sortable key + class byte.
// HBM-bound single pass over 68.5 MB (≈3 us at 23.3 TB/s).
// key = (~orderable(score))<<32 | idx : ascending u64 sort gives
// descending score with stable lower-index-first ties (== lax.top_k).
// ---------------------------------------------------------------
__global__ void prep_kernel(const float* __restrict__ x,
                            unsigned long long* __restrict__ keys,
                            unsigned char* __restrict__ cls8) {
    int gid = blockIdx.x * blockDim.x + threadIdx.x;   // 8*32768 threads
    if (gid >= NBATCH * NPAD) return;
    int b = gid >> 15;            // /32768
    int i = gid & (NPAD - 1);

    float score = -FLT_MAX;
    if (i < NPRED) {
        const float* row = x + ((size_t)b * NPRED + i) * NCH;
        float obj = row[4];
        float best = -1.0f; int bc = 0;
        #pragma unroll 4
        for (int c = 0; c < NCLS; ++c) {
            float v = row[5 + c] * obj;
            if (v > best) { best = v; bc = c; }   // first-max like jnp.argmax
        }
        bool valid = (obj > CONF_THR) && (best > CONF_THR);
        score = valid ? best : -1.0f;
        cls8[(size_t)b * NPRED + i] = (unsigned char)bc;
    }
    unsigned ukey = ~orderable(score);
    keys[(size_t)b * NPAD + i] = ((unsigned long long)ukey << 32) | (unsigned)i;
}

// ---------------------------------------------------------------
// K2: per-batch bitonic sort of 32768 u64 keys, one block per batch,
// ENTIRELY IN LDS: 256 KB key array fits only because gfx1250 has
// 320 KB LDS per WGP. Keys async-staged in (ASYNCcnt); all 120
// compare-exchange passes run at LDS latency instead of L2 round
// trips; only the top-4096 sorted keys are written back (the only
// ones consumed downstream).
// ---------------------------------------------------------------
__global__ void sort_kernel(unsigned long long* __restrict__ keys) {
    __shared__ unsigned long long skey[NPAD];   // 256 KB LDS
    unsigned long long* k = keys + (size_t)blockIdx.x * NPAD;

    // async-stage the batch's keys into LDS (HW stalls issue before
    // the 6-bit ASYNCcnt can overflow, so 32 outstanding/thread is fine)
    for (unsigned t = threadIdx.x; t < (unsigned)NPAD; t += blockDim.x) {
        unsigned lds_addr = (unsigned)(uintptr_t)(&skey[t]);
        const unsigned long long* gp = k + t;
        asm volatile("global_load_async_to_lds_b64 %0, %1, off"
                     :: "v"(lds_addr), "v"(gp) : "memory");
    }
    asm volatile("s_wait_asynccnt 0x0" ::: "memory");
    __syncthreads();

    for (unsigned kk = 2; kk <= (unsigned)NPAD; kk <<= 1) {
        for (unsigned j = kk >> 1; j > 0; j >>= 1) {
            for (unsigned i = threadIdx.x; i < (unsigned)NPAD; i += blockDim.x) {
                unsigned ixj = i ^ j;
                if (ixj > i) {
                    unsigned long long a = skey[i], c = skey[ixj];
                    bool up = ((i & kk) == 0);
                    if ((a > c) == up) { skey[i] = c; skey[ixj] = a; }
                }
            }
            __syncthreads();
        }
    }
    // write back only the top TOPK sorted keys
    for (unsigned t = threadIdx.x; t < (unsigned)TOPK; t += blockDim.x)
        k[t] = skey[t];
}

// ---------------------------------------------------------------
// K3: decode top-4096 per batch -> class-offset boxes (float4),
// det rows [x1,y1,x2,y2,conf,cls], valid bytes.
// ---------------------------------------------------------------
__global__ void gather_kernel(const float* __restrict__ x,
                              const unsigned long long* __restrict__ keys,
                              const unsigned char* __restrict__ cls8,
                              float4* __restrict__ obox,
                              float* __restrict__ det,
                              unsigned char* __restrict__ val8) {
    int gid = blockIdx.x * blockDim.x + threadIdx.x;   // 8*4096 threads
    if (gid >= NBATCH * TOPK) return;
    int b = gid >> 12;
    int r = gid & (TOPK - 1);

    unsigned long long key = keys[(size_t)b * NPAD + r];
    unsigned idx  = (unsigned)(key & 0xFFFFFFFFu);
    unsigned ukey = (unsigned)(key >> 32);
    float top_s = inv_orderable(~ukey);
    bool valid = (top_s > CONF_THR) && (idx < NPRED);

    float x1 = 0.f, y1 = 0.f, x2 = 0.f, y2 = 0.f, cls = 0.f, conf = 0.f;
    if (idx < NPRED) {
        const float* row = x + ((size_t)b * NPRED + idx) * NCH;
        float cx = row[0], cy = row[1], w = row[2], h = row[3];
        x1 = cx - 0.5f * w;  y1 = cy - 0.5f * h;
        x2 = cx + 0.5f * w;  y2 = cy + 0.5f * h;
        cls = (float)cls8[(size_t)b * NPRED + idx];
        conf = top_s;                  // score == conf for valid entries
    }
    float off = cls * CLS_OFF;
    obox[(size_t)b * TOPK + r] = make_float4(x1 + off, y1 + off, x2 + off, y2 + off);
    float* d = det + ((size_t)b * TOPK + r) * 6;
    d[0] = x1; d[1] = y1; d[2] = x2; d[3] = y2; d[4] = conf; d[5] = cls;
    val8[(size_t)b * TOPK + r] = valid ? 1 : 0;
}

// ---------------------------------------------------------------
// K4: greedy NMS scan, one block (256 threads) per batch.
// Stage all 4096 offset boxes (64 KB) into LDS with
// global_load_async_to_lds_b128 (ASYNCcnt). Serial keep/suppress
// loop with parallel IoU marking (LDS atomicOr into 128 words).
// Kept rows stream directly to d_out in keep order (== descending
// score order == top_k(kept_score, 300)); tail zero-filled.
// ---------------------------------------------------------------
__global__ void scan_kernel(const float4* __restrict__ obox,
                            const float* __restrict__ det,
                            const unsigned char* __restrict__ val8,
                            float* __restrict__ out) {
    __shared__ float4        lbox[TOPK];     // 64 KB
    __shared__ unsigned char lval[TOPK];     // 4 KB
    __shared__ unsigned      sup[TOPK / 32]; // 128 words
    __shared__ int s_count;
    __shared__ int s_flag;

    int b = blockIdx.x;
    int tid = threadIdx.x;
    const float4* gsrc = obox + (size_t)b * TOPK;

    // --- async stage boxes into LDS (tracked by ASYNCcnt) ---
    for (int j = tid; j < TOPK; j += 256) {
        unsigned lds_addr = (unsigned)(uintptr_t)(&lbox[j]);
        const float4* gp = gsrc + j;
        asm volatile("global_load_async_to_lds_b128 %0, %1, off"
                     :: "v"(lds_addr), "v"(gp) : "memory");
    }
    // overlap: validity bytes + suppression init while async copies fly
    for (int j = tid; j < TOPK; j += 256)
        lval[j] = val8[(size_t)b * TOPK + j];
    if (tid < TOPK / 32) sup[tid] = 0u;
    if (tid == 0) s_count = 0;
    asm volatile("s_wait_asynccnt 0x0" ::: "memory");
    __syncthreads();

    for (int i = 0; i < TOPK; ++i) {
        if (tid == 0) {
            unsigned w = sup[i >> 5];
            int keep = (lval[i] != 0) && !((w >> (i & 31)) & 1u);
            s_flag = keep;
            if (keep) {
                if (s_count < MAX_DET) {
                    const float* src = det + ((size_t)b * TOPK + i) * 6;
                    float* dst = out + (size_t)b * MAX_DET * 6 + (size_t)s_count * 6;
                    #pragma unroll
                    for (int c = 0; c < 6; ++c) dst[c] = src[c];
                }
                s_count++;
            }
        }
        __syncthreads();
        if (s_flag) {
            float4 bi = lbox[i];
            float areaA = (bi.z - bi.x) * (bi.w - bi.y);
            for (int j = tid; j < TOPK; j += 256) {
                float4 bj = lbox[j];
                float lx = fmaxf(bi.x, bj.x), ly = fmaxf(bi.y, bj.y);
                float rx = fminf(bi.z, bj.z), ry = fminf(bi.w, bj.w);
                float w = fmaxf(rx - lx, 0.f), h = fmaxf(ry - ly, 0.f);
                float inter = w * h;
                float areaB = (bj.z - bj.x) * (bj.w - bj.y);
                float iou = inter / (areaA + areaB - inter + 1e-9f);
                if (iou > IOU_THR) atomicOr(&sup[j >> 5], 1u << (j & 31));
            }
        }
        __syncthreads();
    }

    // zero-fill rows [count, 300): deterministic full-output write
    int z = s_count < MAX_DET ? s_count : MAX_DET;
    int tail = (MAX_DET - z) * 6;
    float* base = out + (size_t)b * MAX_DET * 6 + (size_t)z * 6;
    for (int e = tid; e < tail; e += 256) base[e] = 0.0f;
}

// ---------------------------------------------------------------
extern "C" void kernel_launch(void* const* d_in, const int* in_sizes, int n_in,
                              void* d_out, int out_size, void* d_ws, size_t ws_size,
                              hipStream_t stream) {
    (void)in_sizes; (void)n_in; (void)out_size; (void)ws_size;
    const float* x = (const float*)d_in[0];
    float* out = (float*)d_out;
    char* ws = (char*)d_ws;

    unsigned long long* keys = (unsigned long long*)(ws + WS_KEYS);
    unsigned char*      cls8 = (unsigned char*)(ws + WS_CLS);
    float4*             obox = (float4*)(ws + WS_OBOX);
    float*              det  = (float*)(ws + WS_DET);
    unsigned char*      val8 = (unsigned char*)(ws + WS_VAL);

    // K1: 8*32768 threads, 256/block (8 wave32 per block)
    prep_kernel<<<(NBATCH * NPAD) / 256, 256, 0, stream>>>(x, keys, cls8);
    // K2: one 1024-thread block per batch, 256 KB LDS-resident sort
    sort_kernel<<<NBATCH, 1024, 0, stream>>>(keys);
    // K3: 8*4096 threads
    gather_kernel<<<(NBATCH * TOPK) / 256, 256, 0, stream>>>(x, keys, cls8, obox, det, val8);
    // K4: one block per batch (serial greedy scan)
    scan_kernel<<<NBATCH, 256, 0, stream>>>(obox, det, val8, out);
}